// RNN_Model_77730318123523
// MI455X (gfx1250) — compile-verified
//
#include <hip/hip_runtime.h>
#include <hip/hip_bf16.h>

// CDNA5 / gfx1250: wave32, f32 WMMA 16x16x4.
typedef float v2f __attribute__((ext_vector_type(2)));
typedef float v8f __attribute__((ext_vector_type(8)));

#define VOCAB 512
#define HID   512
#define OUTD  512
#define BATCH 32
#define SEQ   2048
#define MTOT  (BATCH * SEQ)   // 65536 rows
#define WLDA  (VOCAB + HID)   // W_i2h leading dim = 1024

// -------------------------------------------------------------------------
// Phase 1: xproj[m, n] = sum_k emb[x[m], k] * Wx[n, k] + b_i2h[n]
// Grid: one block per 16-row M-tile; 8 waves each own a 64-wide N-slab.
// Embedding gather fused into the A-operand load (emb is L2-resident, 1 MB).
// -------------------------------------------------------------------------
__global__ __launch_bounds__(256) void xproj_kernel(
    const int* __restrict__ x, const float* __restrict__ emb,
    const float* __restrict__ W, const float* __restrict__ bias,
    float* __restrict__ xp)
{
  const int lane = threadIdx.x & 31;
  const int wave = threadIdx.x >> 5;
  const int l15  = lane & 15;       // M for A-operand, N for B-operand
  const int hi   = lane >> 4;       // lane-half selects K-pair
  const int kh   = hi * 2;
  const int m0   = blockIdx.x * 16;
  const int nb   = wave * 64;

  const float* arow = emb + (size_t)x[m0 + l15] * VOCAB;
  const float* bp0 = W + (size_t)(nb +  0 + l15) * WLDA;
  const float* bp1 = W + (size_t)(nb + 16 + l15) * WLDA;
  const float* bp2 = W + (size_t)(nb + 32 + l15) * WLDA;
  const float* bp3 = W + (size_t)(nb + 48 + l15) * WLDA;

  v8f c[4] = {v8f{}, v8f{}, v8f{}, v8f{}};
  #pragma unroll 4
  for (int k = 0; k < VOCAB; k += 4) {
    v2f a  = *(const v2f*)(arow + k + kh);     // A[m][kh:kh+2]
    v2f b0 = *(const v2f*)(bp0  + k + kh);     // Wx[n][kh:kh+2] == B[kh][n]
    v2f b1 = *(const v2f*)(bp1  + k + kh);
    v2f b2 = *(const v2f*)(bp2  + k + kh);
    v2f b3 = *(const v2f*)(bp3  + k + kh);
    c[0] = __builtin_amdgcn_wmma_f32_16x16x4_f32(false, a, false, b0, (short)0, c[0], false, false);
    c[1] = __builtin_amdgcn_wmma_f32_16x16x4_f32(false, a, false, b1, (short)0, c[1], false, false);
    c[2] = __builtin_amdgcn_wmma_f32_16x16x4_f32(false, a, false, b2, (short)0, c[2], false, false);
    c[3] = __builtin_amdgcn_wmma_f32_16x16x4_f32(false, a, false, b3, (short)0, c[3], false, false);
  }
  #pragma unroll
  for (int j = 0; j < 4; ++j) {
    const int n = nb + 16 * j + l15;
    const float bj = bias[n];
    #pragma unroll
    for (int r = 0; r < 8; ++r)   // C/D layout: VGPR r -> row r (+8 for hi half)
      xp[(size_t)(m0 + r + hi * 8) * HID + n] = c[j][r] + bj;
  }
}

// -------------------------------------------------------------------------
// Phase 2: h_t = xproj_t + h_{t-1} @ Wh^T, serial in t, independent per batch.
// 2 workgroups (1024 thr = 32 waves) each own 16 batch rows on one WGP.
// h-state lives in LDS (padded stride -> conflict-free); Wh streams from L2.
// Overwrites xproj in place with h_t (same addresses, same wave).
// -------------------------------------------------------------------------
#define HPAD 516   // 516 % 64 == 4: rows map to distinct banks for 16-row broadcast
__global__ __launch_bounds__(1024) void rnn_kernel(
    const float* __restrict__ W,    // W_i2h; Wh row n = W + n*WLDA + VOCAB
    float* __restrict__ hbuf,       // in: xproj [B,T,HID]; out: h_t in place
    float* __restrict__ hT)
{
  __shared__ float hs[16 * HPAD];   // 33 KB of the 320 KB WGP LDS
  const int lane = threadIdx.x & 31;
  const int wave = threadIdx.x >> 5;
  const int l15  = lane & 15;
  const int hi   = lane >> 4;
  const int kh   = hi * 2;
  const int b0   = blockIdx.x * 16;       // batch slab
  const int n    = wave * 16 + l15;       // this wave's N-tile column
  const float* wrow = W + (size_t)n * WLDA + VOCAB;

  for (int i = threadIdx.x; i < 16 * HPAD; i += blockDim.x) hs[i] = 0.0f;
  __syncthreads();

  v8f c = {};
  for (int t = 0; t < SEQ; ++t) {
    #pragma unroll
    for (int r = 0; r < 8; ++r)           // C := xproj[b, t, n]
      c[r] = hbuf[((size_t)(b0 + r + hi * 8) * SEQ + t) * HID + n];
    #pragma unroll 4
    for (int k = 0; k < HID; k += 4) {
      v2f a = *(const v2f*)(&hs[l15 * HPAD + k + kh]);   // h_{t-1}[m][kh:kh+2]
      v2f b = *(const v2f*)(wrow + k + kh);              // Wh[n][kh:kh+2]
      c = __builtin_amdgcn_wmma_f32_16x16x4_f32(false, a, false, b, (short)0, c, false, false);
    }
    __syncthreads();                      // all waves done reading h_{t-1}
    #pragma unroll
    for (int r = 0; r < 8; ++r) {
      hs[(r + hi * 8) * HPAD + n] = c[r];                            // h_t -> LDS
      hbuf[((size_t)(b0 + r + hi * 8) * SEQ + t) * HID + n] = c[r];  // h_t -> hbuf (overwrites xproj)
    }
    __syncthreads();                      // h_t visible before next step
  }
  #pragma unroll
  for (int r = 0; r < 8; ++r)             // final state hT (32, 512)
    hT[(size_t)(b0 + r + hi * 8) * HID + n] = c[r];
}

// -------------------------------------------------------------------------
// Phase 3: out[m, n] = sum_k h[m, k] * W_fc[n, k] + b_fc[n]
// -------------------------------------------------------------------------
__global__ __launch_bounds__(256) void fc_kernel(
    const float* __restrict__ hsg, const float* __restrict__ W,
    const float* __restrict__ bias, float* __restrict__ out)
{
  const int lane = threadIdx.x & 31;
  const int wave = threadIdx.x >> 5;
  const int l15  = lane & 15;
  const int hi   = lane >> 4;
  const int kh   = hi * 2;
  const int m0   = blockIdx.x * 16;
  const int nb   = wave * 64;

  const float* arow = hsg + (size_t)(m0 + l15) * HID;
  const float* bp0 = W + (size_t)(nb +  0 + l15) * HID;
  const float* bp1 = W + (size_t)(nb + 16 + l15) * HID;
  const float* bp2 = W + (size_t)(nb + 32 + l15) * HID;
  const float* bp3 = W + (size_t)(nb + 48 + l15) * HID;

  v8f c[4] = {v8f{}, v8f{}, v8f{}, v8f{}};
  #pragma unroll 4
  for (int k = 0; k < HID; k += 4) {
    v2f a  = *(const v2f*)(arow + k + kh);
    v2f b0 = *(const v2f*)(bp0  + k + kh);
    v2f b1 = *(const v2f*)(bp1  + k + kh);
    v2f b2 = *(const v2f*)(bp2  + k + kh);
    v2f b3 = *(const v2f*)(bp3  + k + kh);
    c[0] = __builtin_amdgcn_wmma_f32_16x16x4_f32(false, a, false, b0, (short)0, c[0], false, false);
    c[1] = __builtin_amdgcn_wmma_f32_16x16x4_f32(false, a, false, b1, (short)0, c[1], false, false);
    c[2] = __builtin_amdgcn_wmma_f32_16x16x4_f32(false, a, false, b2, (short)0, c[2], false, false);
    c[3] = __builtin_amdgcn_wmma_f32_16x16x4_f32(false, a, false, b3, (short)0, c[3], false, false);
  }
  #pragma unroll
  for (int j = 0; j < 4; ++j) {
    const int n = nb + 16 * j + l15;
    const float bj = bias[n];
    #pragma unroll
    for (int r = 0; r < 8; ++r)
      out[(size_t)(m0 + r + hi * 8) * OUTD + n] = c[j][r] + bj;
  }
}

extern "C" void kernel_launch(void* const* d_in, const int* in_sizes, int n_in,
                              void* d_out, int out_size, void* d_ws, size_t ws_size,
                              hipStream_t stream) {
  const int*   x    = (const int*)  d_in[0];
  const float* emb  = (const float*)d_in[1];
  const float* Wi2h = (const float*)d_in[2];
  const float* bi2h = (const float*)d_in[3];
  const float* Wfc  = (const float*)d_in[4];
  const float* bfc  = (const float*)d_in[5];

  float* out = (float*)d_out;                 // [B*T, OUT] flat
  float* hT  = out + (size_t)MTOT * OUTD;     // tuple tail: [B, HID]
  float* xp  = (float*)d_ws;                  // 128 MiB: xproj, then h_t in place

  xproj_kernel<<<MTOT / 16, 256, 0, stream>>>(x, emb, Wi2h, bi2h, xp);
  rnn_kernel  <<<BATCH / 16, 1024, 0, stream>>>(Wi2h, xp, hT);
  fc_kernel   <<<MTOT / 16, 256, 0, stream>>>(xp, Wfc, bfc, out);
}